// ComplexSSM_18562848653623
// MI455X (gfx1250) — compile-verified
//
#include <hip/hip_runtime.h>
#include <hip/hip_bf16.h>

typedef __attribute__((ext_vector_type(2))) float v2f;
typedef __attribute__((ext_vector_type(8))) float v8f;
typedef __attribute__((ext_vector_type(4))) int   v4i;

#define D_IN   1024
#define R_S    256
#define NB     8
#define NT     2048
#define M_TOTAL (NB * NT)     // 16384 rows (b*T + t)
#define N_TOTAL (3 * R_S)     // 768 cols: [0,256)=u_proj, [256,768)=res
#define E_DIM   (2 * R_S)     // 512

// GEMM tiling: block tile 64x256, 8 waves, each wave 32x64 (c[2][4])
#define BM 64
#define BN 256
#define BK 16
#define LDP 20                // padded LDS row (floats): 80B keeps 16B-aligned
                              // b128 chunks AND stride 20 is bank-conflict-free
#define NSLAB (D_IN / BK)     // 64

// chunked scan
#define CHUNKS 8
#define CLEN   (NT / CHUNKS)  // 256

#if __has_builtin(__builtin_amdgcn_global_load_async_to_lds_b128) && \
    __has_builtin(__builtin_amdgcn_s_wait_asynccnt)
#define USE_ASYNC 1
typedef __attribute__((address_space(1))) v4i gv4i;   // global int4
typedef __attribute__((address_space(3))) v4i lv4i;   // LDS int4
#else
#define USE_ASYNC 0
#endif

// ---------------------------------------------------------------------------
// Kernel 1: fused GEMM  P[m, 0:256]  = u @ W_proj^T ; P[m, 256:768] = u @ W_res^T
// f32 WMMA 16x16x4, double-buffered LDS, async global->LDS b128 staging.
// LDS layout: As[m][k], Bs[n][k] (K-contiguous -> fragment reads are reg pairs)
// ---------------------------------------------------------------------------
__global__ __launch_bounds__(256) void gemm_fused(
    const float* __restrict__ U,      // (M_TOTAL, D_IN)
    const float* __restrict__ Wproj,  // (R_S, D_IN)
    const float* __restrict__ Wres,   // (2*R_S, D_IN)
    float* __restrict__ P)            // (M_TOTAL, N_TOTAL)
{
  __shared__ float As[2][BM][LDP];
  __shared__ float Bs[2][BN][LDP];

  const int tid   = threadIdx.x;
  const int tileM = blockIdx.y * BM;
  const int tileN = blockIdx.x * BN;

  const int w    = tid >> 5;
  const int lane = tid & 31;
  const int half = lane >> 4;      // 0: lanes 0-15, 1: lanes 16-31
  const int lr   = lane & 15;
  const int wm   = (w & 1) * 32;   // wave M offset (0,32)
  const int wn   = (w >> 1) * 64;  // wave N offset (0,64,128,192)

  // --- per-thread staging coordinates (fixed across slabs) ---
  const int aml = tid >> 2;                 // 0..63
  const int akq = (tid & 3) * 4;            // 0,4,8,12
  const float* aG = U + (size_t)(tileM + aml) * D_IN + akq;

  int bnl[4], bkq[4];
  const float* bG[4];
#pragma unroll
  for (int rep = 0; rep < 4; ++rep) {
    const int idx = tid + rep * 256;        // 0..1023
    bnl[rep] = idx >> 2;                    // 0..255
    bkq[rep] = (idx & 3) * 4;
    const int n = tileN + bnl[rep];
    const float* wrow = (n < R_S) ? (Wproj + (size_t)n * D_IN)
                                  : (Wres  + (size_t)(n - R_S) * D_IN);
    bG[rep] = wrow + bkq[rep];
  }

#if USE_ASYNC
#define LOAD_SLAB(K0, BUF)                                                     \
  do {                                                                         \
    __builtin_amdgcn_global_load_async_to_lds_b128(                            \
        (gv4i*)(aG + (K0)), (lv4i*)&As[(BUF)][aml][akq], 0, 0);                \
    __builtin_amdgcn_global_load_async_to_lds_b128(                            \
        (gv4i*)(bG[0] + (K0)), (lv4i*)&Bs[(BUF)][bnl[0]][bkq[0]], 0, 0);       \
    __builtin_amdgcn_global_load_async_to_lds_b128(                            \
        (gv4i*)(bG[1] + (K0)), (lv4i*)&Bs[(BUF)][bnl[1]][bkq[1]], 0, 0);       \
    __builtin_amdgcn_global_load_async_to_lds_b128(                            \
        (gv4i*)(bG[2] + (K0)), (lv4i*)&Bs[(BUF)][bnl[2]][bkq[2]], 0, 0);       \
    __builtin_amdgcn_global_load_async_to_lds_b128(                            \
        (gv4i*)(bG[3] + (K0)), (lv4i*)&Bs[(BUF)][bnl[3]][bkq[3]], 0, 0);       \
  } while (0)
#define WAIT_SLAB() __builtin_amdgcn_s_wait_asynccnt(0)
#else
#define ST4(DST, IDX, V)                                                       \
  do { (DST)[(IDX) + 0] = (V).x; (DST)[(IDX) + 1] = (V).y;                     \
       (DST)[(IDX) + 2] = (V).z; (DST)[(IDX) + 3] = (V).w; } while (0)
#define LOAD_SLAB(K0, BUF)                                                     \
  do {                                                                         \
    const float4 av = *(const float4*)(aG + (K0));                             \
    ST4(As[(BUF)][aml], akq, av);                                              \
    const float4 b0 = *(const float4*)(bG[0] + (K0));                          \
    ST4(Bs[(BUF)][bnl[0]], bkq[0], b0);                                        \
    const float4 b1 = *(const float4*)(bG[1] + (K0));                          \
    ST4(Bs[(BUF)][bnl[1]], bkq[1], b1);                                        \
    const float4 b2 = *(const float4*)(bG[2] + (K0));                          \
    ST4(Bs[(BUF)][bnl[2]], bkq[2], b2);                                        \
    const float4 b3 = *(const float4*)(bG[3] + (K0));                          \
    ST4(Bs[(BUF)][bnl[3]], bkq[3], b3);                                        \
  } while (0)
#define WAIT_SLAB() ((void)0)
#endif

// One BK-deep slab of WMMAs out of LDS buffer BUF.
#define COMPUTE(BUF)                                                           \
  do {                                                                         \
    _Pragma("unroll")                                                          \
    for (int kk = 0; kk < BK; kk += 4) {                                       \
      v2f a[2], b[4];                                                          \
      _Pragma("unroll")                                                        \
      for (int ms = 0; ms < 2; ++ms) {                                         \
        const int row = wm + ms * 16 + lr;                                     \
        a[ms].x = As[(BUF)][row][kk + 2 * half + 0];                           \
        a[ms].y = As[(BUF)][row][kk + 2 * half + 1];                           \
      }                                                                        \
      _Pragma("unroll")                                                        \
      for (int ns = 0; ns < 4; ++ns) {                                         \
        const int col = wn + ns * 16 + lr;                                     \
        b[ns].x = Bs[(BUF)][col][kk + 2 * half + 0];                           \
        b[ns].y = Bs[(BUF)][col][kk + 2 * half + 1];                           \
      }                                                                        \
      _Pragma("unroll")                                                        \
      for (int ms = 0; ms < 2; ++ms)                                           \
        _Pragma("unroll")                                                      \
        for (int ns = 0; ns < 4; ++ns)                                         \
          c[ms][ns] = __builtin_amdgcn_wmma_f32_16x16x4_f32(                   \
              false, a[ms], false, b[ns], (short)0, c[ms][ns], false, false);  \
    }                                                                          \
  } while (0)

  v8f c[2][4] = {};

  LOAD_SLAB(0, 0);
  WAIT_SLAB();
  __syncthreads();

  // branch-free steady state: always prefetch next slab, compute current
  for (int i = 0; i < NSLAB - 1; ++i) {
    const int buf = i & 1;
    LOAD_SLAB((i + 1) * BK, buf ^ 1);
    COMPUTE(buf);
    WAIT_SLAB();        // this wave's copies into buf^1 have landed
    __syncthreads();    // everyone's copies visible; reads of buf done
  }
  COMPUTE((NSLAB - 1) & 1);   // peeled last slab (no prefetch, no barrier)

  // epilogue: C 16x16 f32 layout -> row = v + 8*half, col = lr
#pragma unroll
  for (int ms = 0; ms < 2; ++ms)
#pragma unroll
    for (int ns = 0; ns < 4; ++ns)
#pragma unroll
      for (int v = 0; v < 8; ++v) {
        const int rg = tileM + wm + ms * 16 + v + 8 * half;
        const int cg = tileN + wn + ns * 16 + lr;
        P[(size_t)rg * N_TOTAL + cg] = c[ms][ns][v];
      }
}

// ---------------------------------------------------------------------------
// Chunked parallel scan of h[t] = a*h[t-1] + u_proj[t],  h[-1] = h0.
// ---------------------------------------------------------------------------
__device__ __forceinline__ void decay_params(float lraw, float om,
                                             float& ar, float& ai) {
  const float lam = -5.0f / (1.0f + expf(-lraw));  // -5*sigmoid(lraw)
  const float g   = expf(lam);
  ar = g * cosf(om);
  ai = g * sinf(om);
}

// Phase A: zero-state partial scan per chunk; store end value E_c.
__global__ __launch_bounds__(R_S) void scan_partials(
    const float* __restrict__ lraw, const float* __restrict__ omega,
    const float* __restrict__ P, float2* __restrict__ Carr)
{
  const int b = blockIdx.x >> 3;
  const int c = blockIdx.x & (CHUNKS - 1);
  const int r = threadIdx.x;
  float ar, ai;
  decay_params(lraw[r], omega[r], ar, ai);

  float hr = 0.0f, hi = 0.0f;
  const float* Pb = P + ((size_t)b * NT + (size_t)c * CLEN) * N_TOTAL + r;
  for (int t = 0; t < CLEN; ++t) {
    const float p  = Pb[(size_t)t * N_TOTAL];
    const float nr = ar * hr - ai * hi + p;
    const float ni = ar * hi + ai * hr;
    hr = nr; hi = ni;
  }
  Carr[(size_t)(b * R_S + r) * CHUNKS + c] = make_float2(hr, hi);
}

// Phase B: sequential carry combine: H_c = E_c + a^CLEN * H_{c-1}, H_{-1}=h0.
// Overwrites Carr[c] with the carry-in for chunk c.
__global__ __launch_bounds__(R_S) void scan_combine(
    const float* __restrict__ h0r, const float* __restrict__ h0i,
    const float* __restrict__ lraw, const float* __restrict__ omega,
    float2* __restrict__ Carr)
{
  const int b = blockIdx.x;
  const int r = threadIdx.x;
  float ar, ai;
  decay_params(lraw[r], omega[r], ar, ai);
  // a^CLEN by 8 complex squarings (CLEN = 256) -- avoids huge-arg trig
  float aLr = ar, aLi = ai;
#pragma unroll
  for (int s = 0; s < 8; ++s) {
    const float xr = aLr * aLr - aLi * aLi;
    const float xi = 2.0f * aLr * aLi;
    aLr = xr; aLi = xi;
  }

  float Hr = h0r[b * R_S + r];
  float Hi = h0i[b * R_S + r];
  float2* row = Carr + (size_t)(b * R_S + r) * CHUNKS;
#pragma unroll
  for (int c = 0; c < CHUNKS; ++c) {
    const float2 E = row[c];
    row[c] = make_float2(Hr, Hi);                 // carry-in for chunk c
    const float nr = E.x + aLr * Hr - aLi * Hi;
    const float ni = E.y + aLr * Hi + aLi * Hr;
    Hr = nr; Hi = ni;
  }
}

// Phase C: exact per-chunk scan from carry-in; fuse residual add in place:
// P[m,256+j] <- ssm[j] + res[j]. Last chunk writes final_r / final_i.
__global__ __launch_bounds__(R_S) void scan_apply(
    const float* __restrict__ lraw, const float* __restrict__ omega,
    const float2* __restrict__ Carr, float* __restrict__ P,
    float* __restrict__ final_r, float* __restrict__ final_i)
{
  const int b = blockIdx.x >> 3;
  const int c = blockIdx.x & (CHUNKS - 1);
  const int r = threadIdx.x;
  float ar, ai;
  decay_params(lraw[r], omega[r], ar, ai);

  const float2 h0c = Carr[(size_t)(b * R_S + r) * CHUNKS + c];
  float hr = h0c.x, hi = h0c.y;

  float* Pb = P + ((size_t)b * NT + (size_t)c * CLEN) * N_TOTAL;
  for (int t = 0; t < CLEN; ++t) {
    float* row = Pb + (size_t)t * N_TOTAL;
    const float p  = row[r];
    const float nr = ar * hr - ai * hi + p;
    const float ni = ar * hi + ai * hr;
    hr = nr; hi = ni;
    row[R_S + r]     += hr;   // ssm_out[..,r]     + res[..,r]
    row[2 * R_S + r] += hi;   // ssm_out[..,256+r] + res[..,256+r]
  }
  if (c == CHUNKS - 1) {
    final_r[b * R_S + r] = hr;
    final_i[b * R_S + r] = hi;
  }
}

// ---------------------------------------------------------------------------
// Kernel 3: LayerNorm over 512-wide rows of P[:,256:768] -> out.
// ---------------------------------------------------------------------------
__global__ __launch_bounds__(256) void ln_kernel(
    const float* __restrict__ P, const float* __restrict__ gamma,
    const float* __restrict__ beta, float* __restrict__ out)
{
  const int w    = threadIdx.x >> 5;
  const int lane = threadIdx.x & 31;
  const int m    = blockIdx.x * 8 + w;

  const float* x = P + (size_t)m * N_TOTAL + R_S;  // 512 pre-LN values
  float v[16];
  float s = 0.0f;
#pragma unroll
  for (int i = 0; i < 16; ++i) { v[i] = x[lane + 32 * i]; s += v[i]; }
#pragma unroll
  for (int off = 16; off; off >>= 1) s += __shfl_xor(s, off, 32);
  const float mu = s * (1.0f / 512.0f);

  float q = 0.0f;
#pragma unroll
  for (int i = 0; i < 16; ++i) { const float d = v[i] - mu; q += d * d; }
#pragma unroll
  for (int off = 16; off; off >>= 1) q += __shfl_xor(q, off, 32);
  const float rs = rsqrtf(q * (1.0f / 512.0f) + 1e-5f);

  float* o = out + (size_t)m * E_DIM;
#pragma unroll
  for (int i = 0; i < 16; ++i) {
    const int j = lane + 32 * i;
    o[j] = (v[i] - mu) * rs * gamma[j] + beta[j];
  }
}

// ---------------------------------------------------------------------------
extern "C" void kernel_launch(void* const* d_in, const int* in_sizes, int n_in,
                              void* d_out, int out_size, void* d_ws,
                              size_t ws_size, hipStream_t stream) {
  const float* u     = (const float*)d_in[0];
  const float* h0r   = (const float*)d_in[1];
  const float* h0i   = (const float*)d_in[2];
  const float* lraw  = (const float*)d_in[3];
  const float* omega = (const float*)d_in[4];
  const float* Wproj = (const float*)d_in[5];
  const float* Wres  = (const float*)d_in[6];
  const float* gamma = (const float*)d_in[7];
  const float* beta  = (const float*)d_in[8];

  float* out = (float*)d_out;
  float* P   = (float*)d_ws;                               // 16384 x 768 f32
  float2* Carr = (float2*)(P + (size_t)M_TOTAL * N_TOTAL); // (B*R, CHUNKS)

  dim3 g1(N_TOTAL / BN, M_TOTAL / BM);  // (3, 256)
  gemm_fused<<<g1, 256, 0, stream>>>(u, Wproj, Wres, P);

  float* finr = out + (size_t)M_TOTAL * E_DIM;   // (8,256)
  float* fini = finr + NB * R_S;                 // (8,256)

  scan_partials<<<NB * CHUNKS, R_S, 0, stream>>>(lraw, omega, P, Carr);
  scan_combine<<<NB, R_S, 0, stream>>>(h0r, h0i, lraw, omega, Carr);
  scan_apply<<<NB * CHUNKS, R_S, 0, stream>>>(lraw, omega, Carr, P, finr, fini);

  ln_kernel<<<M_TOTAL / 8, 256, 0, stream>>>(P, gamma, beta, out);
}